// MultiEdgeDenseGCNConv_49117245997762
// MI455X (gfx1250) — compile-verified
//
#include <hip/hip_runtime.h>
#include <hip/hip_bf16.h>

// MultiEdgeDenseGCNConv on MI455X (gfx1250, wave32).
// B=16, N=1024, E=3, IN_C=OUT_C=32.
// Main einsum = 48 GEMMs of [1024x1024]x[1024x32] fp32 -> v_wmma_f32_16x16x4_f32.

typedef float v2f __attribute__((ext_vector_type(2)));
typedef float v8f __attribute__((ext_vector_type(8)));

#define GCN_B 16
#define GCN_N 1024
#define GCN_E 3
#define GCN_C 32
#define GCN_K 32          // IN_C
#define GCN_COLS 96       // OUT_C * E

// ---------------------------------------------------------------------------
// Kernel 1: deg_inv_sqrt[b,n,e] = rsqrt(max(sum_m adj[b,n,m,e], 1))
// One block per (b,n) row: the 1024*3 floats adj[b,n,:,:] are contiguous.
// ---------------------------------------------------------------------------
__global__ __launch_bounds__(256)
void deg_kernel(const float* __restrict__ adj, float* __restrict__ dis)
{
    __shared__ float red[3][256];
    const int row = blockIdx.x;                       // b*N + n
    const float* p = adj + (size_t)row * (GCN_N * GCN_E);
    const int tid = threadIdx.x;

    float a0 = 0.f, a1 = 0.f, a2 = 0.f;
    for (int j = tid; j < GCN_N * GCN_E; j += 256) {
        float v = p[j];
        int e = j % 3;
        if (e == 0)      a0 += v;
        else if (e == 1) a1 += v;
        else             a2 += v;
    }
    red[0][tid] = a0; red[1][tid] = a1; red[2][tid] = a2;
    __syncthreads();
    for (int s = 128; s > 0; s >>= 1) {
        if (tid < s) {
            red[0][tid] += red[0][tid + s];
            red[1][tid] += red[1][tid + s];
            red[2][tid] += red[2][tid + s];
        }
        __syncthreads();
    }
    if (tid < 3) {
        float s = fmaxf(red[tid][0], 1.0f);
        dis[(size_t)row * 3 + tid] = rsqrtf(s);
    }
}

// ---------------------------------------------------------------------------
// Kernel 2: T[b,e,m,c] = dis[b,m,e] * (x[b,m,:] @ W[:, e*32+c])
//           S[b,n,c]   = sum_e (x[b,n,:] @ W0[:, e*32+c] + b0[e*32+c])
// One block per 16 rows; W/W0/b0 and the x tile staged in LDS.
// ---------------------------------------------------------------------------
__global__ __launch_bounds__(256)
void lin_kernel(const float* __restrict__ x,  const float* __restrict__ W,
                const float* __restrict__ W0, const float* __restrict__ b0,
                const float* __restrict__ dis,
                float* __restrict__ T, float* __restrict__ S)
{
    __shared__ float xs[16 * GCN_K];
    __shared__ float Ws[GCN_K * GCN_COLS];
    __shared__ float W0s[GCN_K * GCN_COLS];
    __shared__ float b0s[GCN_COLS];

    const int tid = threadIdx.x;
    const int r0  = blockIdx.x * 16;                  // global row b*N + n

    for (int i = tid; i < GCN_K * GCN_COLS; i += 256) { Ws[i] = W[i]; W0s[i] = W0[i]; }
    if (tid < GCN_COLS) b0s[tid] = b0[tid];
    for (int i = tid; i < 16 * GCN_K; i += 256) xs[i] = x[(size_t)r0 * GCN_K + i];
    __syncthreads();

    // T (column-normalized per-edge-type transform)
    for (int idx = tid; idx < 16 * GCN_COLS; idx += 256) {
        int rl = idx / GCN_COLS, col = idx % GCN_COLS;
        int e = col / GCN_C, c = col % GCN_C;
        float dot = 0.f;
        #pragma unroll
        for (int k = 0; k < GCN_K; ++k) dot += xs[rl * GCN_K + k] * Ws[k * GCN_COLS + col];
        int row = r0 + rl;
        int b = row >> 10, n = row & (GCN_N - 1);
        T[((size_t)(b * GCN_E + e) * GCN_N + n) * GCN_C + c] =
            dis[(size_t)row * 3 + e] * dot;
    }
    // S (self-loop term, pre-summed over edge types)
    for (int idx = tid; idx < 16 * GCN_C; idx += 256) {
        int rl = idx / GCN_C, c = idx % GCN_C;
        float acc = 0.f;
        #pragma unroll
        for (int e = 0; e < GCN_E; ++e) {
            int col = e * GCN_C + c;
            float dot = b0s[col];
            #pragma unroll
            for (int k = 0; k < GCN_K; ++k) dot += xs[rl * GCN_K + k] * W0s[k * GCN_COLS + col];
            acc += dot;
        }
        S[(size_t)(r0 + rl) * GCN_C + c] = acc;
    }
}

// ---------------------------------------------------------------------------
// Kernel 3: out[b,n,c] = mask[b,n] * ( S[b,n,c]
//              + sum_e dis[b,n,e] * sum_m adj[b,n,m,e] * T[b,e,m,c] )
// Block = 128 threads = 4 waves. Block owns a 16-row n-tile of batch b;
// wave w covers m in [w*256, w*256+256) with 64 steps of
// v_wmma_f32_16x16x4_f32 (x3 edge types x2 column halves).
// A fragment (16x4 f32): lanes 0-15 = M rows, regs = K pair sel by lane[4].
// B fragment (4x16 f32): lanes = N cols, regs = K pair sel by lane[4].
// ---------------------------------------------------------------------------
__global__ __launch_bounds__(128)
void gcn_agg_kernel(const float* __restrict__ adj,
                    const float* __restrict__ dis,
                    const float* __restrict__ T,
                    const float* __restrict__ S,
                    const unsigned char* __restrict__ mask,
                    float* __restrict__ out)
{
    constexpr int N = GCN_N, E = GCN_E, C = GCN_C;
    constexpr int NT = N / 16;                        // 64 n-tiles per batch
    __shared__ float red[4 * 512];

    const int blk  = blockIdx.x;
    const int b    = blk / NT;
    const int n0   = (blk % NT) * 16;
    const int tid  = threadIdx.x;
    const int wave = tid >> 5;
    const int lane = tid & 31;
    const int lo   = lane & 15;                       // A: M row / B,C,D: N col
    const int hi   = lane >> 4;                       // selects K pair (and M+8 in C/D)
    const int kk   = hi * 2;

    // This lane's adj row: adj[b, n0+lo, :, :] -- 3072 contiguous floats.
    const float* adjRow = adj + (size_t)(b * N + n0 + lo) * (N * E);
    const float* Tb     = T + (size_t)b * E * N * C;

    v8f acc[E][2];
    #pragma unroll
    for (int e = 0; e < E; ++e)
        #pragma unroll
        for (int h = 0; h < 2; ++h)
            #pragma unroll
            for (int r = 0; r < 8; ++r) acc[e][h][r] = 0.0f;

    const int mbase = wave * 256;
    for (int step = 0; step < 64; ++step) {
        const int m = mbase + step * 4;
        // 6 consecutive floats: (m+kk, e=0..2), (m+kk+1, e=0..2); 8B aligned.
        const float* ap = adjRow + (size_t)(m + kk) * E;
        __builtin_prefetch(adjRow + (size_t)(m + 256) * E, 0, 1);  // global_prefetch_b8
        v2f p0 = *(const v2f*)(ap + 0);
        v2f p1 = *(const v2f*)(ap + 2);
        v2f p2 = *(const v2f*)(ap + 4);
        v2f a[3];
        a[0].x = p0.x; a[0].y = p1.y;                 // e=0: (m+kk,0),(m+kk+1,0)
        a[1].x = p0.y; a[1].y = p2.x;                 // e=1
        a[2].x = p1.x; a[2].y = p2.y;                 // e=2

        #pragma unroll
        for (int e = 0; e < E; ++e) {
            const float* tb = Tb + (size_t)e * N * C + (size_t)(m + kk) * C;
            #pragma unroll
            for (int h = 0; h < 2; ++h) {
                v2f bv;
                bv.x = tb[h * 16 + lo];               // K = m+kk   row of T
                bv.y = tb[C + h * 16 + lo];           // K = m+kk+1 row of T
                acc[e][h] = __builtin_amdgcn_wmma_f32_16x16x4_f32(
                    false, a[e], false, bv, (short)0, acc[e][h], false, false);
            }
        }
    }

    // Apply row normalization dis[b,n,e] to this wave's partial, stash in LDS.
    #pragma unroll
    for (int h = 0; h < 2; ++h) {
        #pragma unroll
        for (int r = 0; r < 8; ++r) {
            const int nl = r + 8 * hi;                // C/D layout: M = r (+8 for hi half)
            float v = 0.f;
            #pragma unroll
            for (int e = 0; e < E; ++e)
                v += dis[(size_t)(b * N + n0 + nl) * E + e] * acc[e][h][r];
            red[wave * 512 + nl * C + h * 16 + lo] = v;
        }
    }
    __syncthreads();

    // Combine 4 K-slices, add self term, apply mask, single write to out.
    #pragma unroll
    for (int it = 0; it < 4; ++it) {
        const int idx = it * 128 + tid;               // 0..511 tile element
        const int nl = idx >> 5, c = idx & 31;
        float s = red[idx] + red[512 + idx] + red[1024 + idx] + red[1536 + idx];
        const size_t row = (size_t)b * N + n0 + nl;
        s += S[row * C + c];
        s *= mask[row] ? 1.0f : 0.0f;                 // numpy bool -> 1 byte
        out[row * C + c] = s;
    }
}

// ---------------------------------------------------------------------------
// Host launcher. Inputs (setup_inputs order): x, adj, mask, W, W0, b0.
// Workspace layout (floats): dis[B*N*E] | T[B*E*N*C] | S[B*N*C]  (~8.6 MB)
// ---------------------------------------------------------------------------
extern "C" void kernel_launch(void* const* d_in, const int* in_sizes, int n_in,
                              void* d_out, int out_size, void* d_ws, size_t ws_size,
                              hipStream_t stream)
{
    (void)in_sizes; (void)n_in; (void)out_size; (void)ws_size;

    const float*         x    = (const float*)d_in[0];
    const float*         adj  = (const float*)d_in[1];
    const unsigned char* mask = (const unsigned char*)d_in[2];
    const float*         W    = (const float*)d_in[3];
    const float*         W0   = (const float*)d_in[4];
    const float*         b0   = (const float*)d_in[5];
    float*               out  = (float*)d_out;

    float* wsf = (float*)d_ws;
    float* dis = wsf;                                              // 16*1024*3
    float* T   = wsf + (size_t)GCN_B * GCN_N * GCN_E;              // 16*3*1024*32
    float* S   = T   + (size_t)GCN_B * GCN_E * GCN_N * GCN_C;      // 16*1024*32

    deg_kernel<<<GCN_B * GCN_N, 256, 0, stream>>>(adj, dis);
    lin_kernel<<<GCN_B * GCN_N / 16, 256, 0, stream>>>(x, W, W0, b0, dis, T, S);
    gcn_agg_kernel<<<GCN_B * (GCN_N / 16), 128, 0, stream>>>(adj, dis, T, S, mask, out);
}